// FCAlwaysLayer_82205674045457
// MI455X (gfx1250) — compile-verified
//
#include <hip/hip_runtime.h>
#include <stdint.h>

// relu(beta - rowsum(W) + x @ W^T)  ==  relu(beta + (x-1) @ W^T)
// Single fp32 WMMA GEMM, TDM async tile DMA, double-buffered LDS.

#define B_DIM 4096
#define I_DIM 4096
#define O_DIM 4096
#define KTILE 16
#define NKT   (I_DIM / KTILE)   // 256 K-chunks
#define ROWP  20                // padded LDS row stride in dwords (16 data + 4 pad)

typedef __attribute__((ext_vector_type(4))) unsigned int u32x4;
typedef __attribute__((ext_vector_type(8))) int          i32x8;
typedef __attribute__((ext_vector_type(4))) int          i32x4;
typedef __attribute__((ext_vector_type(2))) float        v2f;
typedef __attribute__((ext_vector_type(8))) float        v8f;

// TDM: DMA a 128-row x 16-col fp32 tile (tensor row stride 4096 elems) into LDS,
// padding 4 dwords after every 16 dwords so LDS rows land at stride ROWP=20.
__device__ __forceinline__ void tdm_tile_load(const float* gptr, uint32_t lds_off) {
  uint64_t ga = (uint64_t)(uintptr_t)gptr;
  // Group 0: count=1 | lds_addr | global_addr[56:0] | type=2
  u32x4 g0 = { 1u,
               lds_off,
               (uint32_t)ga,
               (uint32_t)(ga >> 32) | (2u << 30) };
  // Group 1: data_size=4B, pad_enable, pad_interval=16dw (code 3), pad_amount=4dw (code 3)
  //          tensor_dim0 = tensor_dim1 = 4096, tile_dim0=16, tile_dim1=128, stride0=4096
  i32x8 g1 = { (int)((2u << 16) | (1u << 20) | (3u << 22) | (3u << 25)),
               (int)(((uint32_t)I_DIM & 0xFFFFu) << 16),                 // dim0[15:0] -> [31:16]
               (int)(((uint32_t)I_DIM >> 16) |
                     (((uint32_t)I_DIM & 0xFFFFu) << 16)),               // dim0[31:16] | dim1[15:0]
               (int)(((uint32_t)I_DIM >> 16) | ((uint32_t)KTILE << 16)), // dim1[31:16] | tile_dim0
               (int)(128u),                                              // tile_dim1=128, tile_dim2=0
               (int)(I_DIM),                                             // tensor_dim0_stride lo32
               0, 0 };
  i32x4 z4 = { 0, 0, 0, 0 };
  i32x8 z8 = { 0, 0, 0, 0, 0, 0, 0, 0 };
  __builtin_amdgcn_tensor_load_to_lds(g0, g1, z4, z4, z8, 0);
}

__global__ __launch_bounds__(256)
void fc_gemm_tdm_wmma(const float* __restrict__ x, const float* __restrict__ W,
                      const float* __restrict__ beta, float* __restrict__ out) {
  __shared__ float As[2][128 * ROWP];
  __shared__ float Bs[2][128 * ROWP];

  const int lane  = threadIdx.x & 31;
  const int wave  = threadIdx.x >> 5;
  const int waveM = wave >> 2;      // 0..1 -> 64-row band
  const int waveN = wave & 3;       // 0..3 -> 32-col band
  const int lmod  = lane & 15;
  const int lhalf = lane >> 4;

  const int tileM = blockIdx.x * 128;
  const int tileN = blockIdx.y * 128;

  const float* Abase = x + (size_t)tileM * I_DIM;
  const float* Bbase = W + (size_t)tileN * I_DIM;

  const uint32_t ldsA0 = (uint32_t)(uintptr_t)&As[0][0];
  const uint32_t ldsA1 = (uint32_t)(uintptr_t)&As[1][0];
  const uint32_t ldsB0 = (uint32_t)(uintptr_t)&Bs[0][0];
  const uint32_t ldsB1 = (uint32_t)(uintptr_t)&Bs[1][0];

  v8f acc[4][2];
  #pragma unroll
  for (int mi = 0; mi < 4; ++mi)
    #pragma unroll
    for (int ni = 0; ni < 2; ++ni)
      acc[mi][ni] = (v8f){0.f,0.f,0.f,0.f,0.f,0.f,0.f,0.f};

  // Prologue: stage 0 into buffer 0 (wave 0 drives the TDM)
  if (wave == 0) {
    tdm_tile_load(Abase, ldsA0);
    tdm_tile_load(Bbase, ldsB0);
  }

  for (int kt = 0; kt < NKT; ++kt) {
    const int buf = kt & 1;
    if (wave == 0) {
      if (kt + 1 < NKT) {
        const float* an = Abase + (size_t)(kt + 1) * KTILE;
        const float* bn = Bbase + (size_t)(kt + 1) * KTILE;
        tdm_tile_load(an, buf ? ldsA0 : ldsA1);
        tdm_tile_load(bn, buf ? ldsB0 : ldsB1);
        __builtin_amdgcn_s_wait_tensorcnt(2);   // pair kt complete; pair kt+1 in flight
      } else {
        __builtin_amdgcn_s_wait_tensorcnt(0);   // last pair complete
      }
    }
    __syncthreads();   // publish stage `buf` to all waves

    const float* Ab = &As[buf][0];
    const float* Bb = &Bs[buf][0];
    #pragma unroll
    for (int k4 = 0; k4 < KTILE / 4; ++k4) {
      const int kk = k4 * 4 + lhalf * 2;        // K = kk, kk+1 for VGPR 0,1
      v2f a[4], b[2];
      #pragma unroll
      for (int mi = 0; mi < 4; ++mi) {          // A frag: row = M, cols = K (fold x-1 here)
        const int r = waveM * 64 + mi * 16 + lmod;
        v2f t = *(const v2f*)(Ab + r * ROWP + kk);
        a[mi] = t - 1.0f;
      }
      #pragma unroll
      for (int ni = 0; ni < 2; ++ni) {          // B frag: row = N (W is [O,I]), cols = K
        const int r = waveN * 32 + ni * 16 + lmod;
        b[ni] = *(const v2f*)(Bb + r * ROWP + kk);
      }
      #pragma unroll
      for (int mi = 0; mi < 4; ++mi)
        #pragma unroll
        for (int ni = 0; ni < 2; ++ni)
          acc[mi][ni] = __builtin_amdgcn_wmma_f32_16x16x4_f32(
              false, a[mi], false, b[ni], (short)0, acc[mi][ni], false, false);
    }
    __syncthreads();   // stage fully consumed before TDM overwrites it
  }

  // Epilogue: + beta, ReLU.  C layout: VGPR r -> M = r + 8*(lane/16), N = lane%16.
  const float betaV = beta[0];
  #pragma unroll
  for (int mi = 0; mi < 4; ++mi) {
    #pragma unroll
    for (int ni = 0; ni < 2; ++ni) {
      const int col = tileN + waveN * 32 + ni * 16 + lmod;
      #pragma unroll
      for (int r = 0; r < 8; ++r) {
        const int row = tileM + waveM * 64 + mi * 16 + lhalf * 8 + r;
        float v = acc[mi][ni][r] + betaV;
        out[(size_t)row * O_DIM + col] = v > 0.0f ? v : 0.0f;
      }
    }
  }
}

extern "C" void kernel_launch(void* const* d_in, const int* in_sizes, int n_in,
                              void* d_out, int out_size, void* d_ws, size_t ws_size,
                              hipStream_t stream) {
  (void)in_sizes; (void)n_in; (void)d_ws; (void)ws_size; (void)out_size;
  const float* x    = (const float*)d_in[0];
  const float* W    = (const float*)d_in[1];
  const float* beta = (const float*)d_in[2];
  float* out        = (float*)d_out;

  dim3 grid(B_DIM / 128, O_DIM / 128, 1);   // 32 x 32 tiles
  dim3 block(256, 1, 1);                    // 8 wave32 waves
  fc_gemm_tdm_wmma<<<grid, block, 0, stream>>>(x, W, beta, out);
}